// S4DFFT_1039382086401
// MI455X (gfx1250) — compile-verified
//
#include <hip/hip_runtime.h>
#include <hip/hip_bf16.h>
#include <math.h>

// ---------------------------------------------------------------------------
// S4D long-conv:  out = ( scan( x @ W_in^T ) ) @ W_out^T
//   B=4, T=8192, D=1024, HALF=32.  All f32.
//
// Memory-bound: ~134 MB streaming read of x + ~134 MB streaming write of out
// + 8 MB L2-resident intermediates  ->  ~12 us floor at 23.3 TB/s.
// Full f32 precision via V_WMMA_F32_16X16X4_F32 (compute is free here).
// Streaming traffic uses non-temporal hints so the 8 MB of intermediates
// stay resident in the 192 MB L2 across the three kernels.
// ---------------------------------------------------------------------------

#define BB    4
#define TT    8192
#define DD    1024
#define HALF  32
#define BT    (BB * TT)          // 32768 rows

typedef float v2f __attribute__((ext_vector_type(2)));
typedef float v8f __attribute__((ext_vector_type(8)));

// ---------------------------------------------------------------------------
// GEMM1: x_proj[BT x 32] = x[BT x 1024] * W_in^T  (W_in is [32 x 1024])
// One wave per 16-row M-tile; 2 N-tiles (32 cols) per wave; K-loop step 4.
// A frag: lane L holds row (L&15), K-pair (k0 + 2*(L>>4)) -> b64 NT load.
// B frag: lane L holds col (L&15), same K-pair; W_in row-major over K -> b64.
// ---------------------------------------------------------------------------
__global__ __launch_bounds__(256) void s4d_gemm1(const float* __restrict__ x,
                                                 const float* __restrict__ Win,
                                                 float* __restrict__ xproj) {
    const int wave = threadIdx.x >> 5;
    const int lane = threadIdx.x & 31;
    const int mtile = blockIdx.x * 8 + wave;        // 2048 tiles / 256 blocks
    const int hsel  = lane >> 4;
    const int l16   = lane & 15;

    const float* ap  = x   + (size_t)(mtile * 16 + l16) * DD + 2 * hsel;
    const float* bp0 = Win + (size_t)l16 * DD + 2 * hsel;
    const float* bp1 = bp0 + (size_t)16 * DD;

    v8f c0 = {}; v8f c1 = {};
    for (int k = 0; k < DD; k += 4) {
        // x is streamed exactly once: non-temporal load (TH=NT).
        v2f a  = __builtin_nontemporal_load((const v2f*)(ap + k));
        v2f b0 = *(const v2f*)(bp0 + k);
        v2f b1 = *(const v2f*)(bp1 + k);
        c0 = __builtin_amdgcn_wmma_f32_16x16x4_f32(false, a, false, b0,
                                                   (short)0, c0, false, false);
        c1 = __builtin_amdgcn_wmma_f32_16x16x4_f32(false, a, false, b1,
                                                   (short)0, c1, false, false);
    }

    // D layout: VGPR r -> row r (lanes 0-15) / row r+8 (lanes 16-31), N=lane%16
    // x_proj is re-read by the scan kernel: keep regular-temporal (L2 resident).
    #pragma unroll
    for (int r = 0; r < 8; ++r) {
        const int grow = mtile * 16 + r + 8 * hsel;
        xproj[(size_t)grow * HALF + l16]      = c0[r];
        xproj[(size_t)grow * HALF + l16 + 16] = c1[r];
    }
}

// ---------------------------------------------------------------------------
// Blocked parallel scan of h_t = lam * h_{t-1} + u_t,  y_t = g * Re(h_t).
// One wave per (b, n) sequence (128 total); each lane owns 256 timesteps.
// Phase 1: local state. Phase 2: Kogge-Stone affine scan over lanes
// (per-segment multiplier lam^256 built by 8 complex squarings).
// Phase 3: replay with carry-in, emit y.
// ---------------------------------------------------------------------------
__global__ __launch_bounds__(32) void s4d_scan(const float* __restrict__ xproj,
                                               const float* __restrict__ log_tau,
                                               const float* __restrict__ freq,
                                               const float* __restrict__ Bp,
                                               const float* __restrict__ Cp,
                                               const float* __restrict__ log_dt,
                                               float* __restrict__ y) {
    const int id   = blockIdx.x;       // 0..127
    const int b    = id >> 5;
    const int n    = id & 31;
    const int lane = threadIdx.x;      // 0..31
    const int SEG  = TT / 32;          // 256

    const float dt    = expf(log_dt[0]);
    const float tau   = fmaxf(expf(log_tau[n]), 1e-4f);
    const float ang   = freq[n] * dt;
    const float lmag  = expf(-tau * dt);
    const float lr    = lmag * cosf(ang);
    const float li    = lmag * sinf(ang);
    const float g     = (fabsf(Bp[n]) + 1e-9f) * (fabsf(Cp[n]) + 1e-9f);

    const float* u = xproj + ((size_t)b * TT + (size_t)lane * SEG) * HALF + n;

    // phase 1: local end-state with zero init
    float hr = 0.f, hi = 0.f;
    #pragma unroll 4
    for (int j = 0; j < SEG; ++j) {
        const float uv = u[(size_t)j * HALF];
        const float nr = fmaf(lr, hr, fmaf(-li, hi, uv));
        const float ni = fmaf(lr, hi, li * hr);
        hr = nr; hi = ni;
    }

    // per-segment multiplier A = lam^256 via 8 complex squarings
    float Ar = lr, Ai = li;
    #pragma unroll
    for (int s = 0; s < 8; ++s) {
        const float nr = Ar * Ar - Ai * Ai;
        const float ni = 2.f * Ar * Ai;
        Ar = nr; Ai = ni;
    }

    // phase 2: inclusive scan  S_l = sum_{l'<=l} A^(l-l') * s_{l'}
    float sr = hr, si = hi;
    float mr = Ar, mi = Ai;
    #pragma unroll
    for (int off = 1; off < 32; off <<= 1) {
        const float pr = __shfl_up(sr, off, 32);
        const float pi = __shfl_up(si, off, 32);
        if (lane >= off) {
            sr += mr * pr - mi * pi;
            si += mr * pi + mi * pr;
        }
        const float nmr = mr * mr - mi * mi;
        const float nmi = 2.f * mr * mi;
        mr = nmr; mi = nmi;
    }

    // exclusive carry-in
    float cr = __shfl_up(sr, 1, 32);
    float ci = __shfl_up(si, 1, 32);
    if (lane == 0) { cr = 0.f; ci = 0.f; }

    // phase 3: replay with carry, write y = g * Re(h)
    hr = cr; hi = ci;
    float* yo = y + ((size_t)b * TT + (size_t)lane * SEG) * HALF + n;
    #pragma unroll 4
    for (int j = 0; j < SEG; ++j) {
        const float uv = u[(size_t)j * HALF];
        const float nr = fmaf(lr, hr, fmaf(-li, hi, uv));
        const float ni = fmaf(lr, hi, li * hr);
        hr = nr; hi = ni;
        yo[(size_t)j * HALF] = g * hr;   // y re-read by GEMM2: regular-temporal
    }
}

// ---------------------------------------------------------------------------
// GEMM2: out[BT x 1024] = y[BT x 32] * W_out^T   (W_out is [1024 x 32])
// grid (2048 mtiles, 8); wave w of block handles ntile = blockIdx.y*8 + w.
// K = 32 -> 8 wmma per tile. A tile (16x32, 2KB) is L1-resident across
// the 8 waves sharing the same mtile. Output stores are non-temporal:
// out is written once and never re-read by the GPU.
// ---------------------------------------------------------------------------
__global__ __launch_bounds__(256) void s4d_gemm2(const float* __restrict__ y,
                                                 const float* __restrict__ Wout,
                                                 float* __restrict__ out) {
    const int wave  = threadIdx.x >> 5;
    const int lane  = threadIdx.x & 31;
    const int mtile = blockIdx.x;                   // 0..2047
    const int ntile = blockIdx.y * 8 + wave;        // 0..63
    const int hsel  = lane >> 4;
    const int l16   = lane & 15;

    const float* ap = y    + (size_t)(mtile * 16 + l16) * HALF + 2 * hsel;
    const float* bp = Wout + (size_t)(ntile * 16 + l16) * HALF + 2 * hsel;

    v8f c = {};
    #pragma unroll
    for (int k = 0; k < HALF; k += 4) {
        v2f a = *(const v2f*)(ap + k);
        v2f b = *(const v2f*)(bp + k);
        c = __builtin_amdgcn_wmma_f32_16x16x4_f32(false, a, false, b,
                                                  (short)0, c, false, false);
    }

    const int col = ntile * 16 + l16;
    #pragma unroll
    for (int r = 0; r < 8; ++r) {
        const int grow = mtile * 16 + r + 8 * hsel;
        __builtin_nontemporal_store(c[r], out + (size_t)grow * DD + col);
    }
}

// ---------------------------------------------------------------------------
extern "C" void kernel_launch(void* const* d_in, const int* in_sizes, int n_in,
                              void* d_out, int out_size, void* d_ws, size_t ws_size,
                              hipStream_t stream) {
    (void)in_sizes; (void)n_in; (void)out_size; (void)ws_size;

    const float* x       = (const float*)d_in[0];
    const float* log_tau = (const float*)d_in[1];
    const float* freq    = (const float*)d_in[2];
    const float* Bp      = (const float*)d_in[3];
    const float* Cp      = (const float*)d_in[4];
    const float* log_dt  = (const float*)d_in[5];
    const float* Win     = (const float*)d_in[6];
    const float* Wout    = (const float*)d_in[7];
    float* out = (float*)d_out;

    float* xproj = (float*)d_ws;                       // 32768*32 f32 = 4 MB
    float* ybuf  = xproj + (size_t)BT * HALF;          // 4 MB

    s4d_gemm1<<<dim3(BT / 16 / 8), dim3(256), 0, stream>>>(x, Win, xproj);
    s4d_scan <<<dim3(BB * HALF),   dim3(32),  0, stream>>>(xproj, log_tau, freq,
                                                           Bp, Cp, log_dt, ybuf);
    s4d_gemm2<<<dim3(BT / 16, 8),  dim3(256), 0, stream>>>(ybuf, Wout, out);
}